// ProjectedGaussianRasterizer_7421703487872
// MI455X (gfx1250) — compile-verified
//
#include <hip/hip_runtime.h>

#define NG      512
#define IMG_H   256
#define IMG_W   256
#define NPIX    (IMG_H * IMG_W)
#define ALPHA_T     (1.0f / 255.0f)
#define TRANS_T     1e-4f
#define ALPHA_CLAMP 0.999f

typedef float v2f __attribute__((ext_vector_type(2)));
typedef float v8f __attribute__((ext_vector_type(8)));

// ws layout (floats); slot NG is a dummy "null gaussian" (op=0 -> alpha==0):
//   [g*8 + 0..5]         : mx, my, A, B, C, op    g in [0, NG]
//   [COL_BASE + g*3 + c] : color                  g in [0, NG]
#define PAR_STRIDE 8
#define COL_BASE   ((NG + 1) * PAR_STRIDE)          // 4104
#define WS_FLOATS  (COL_BASE + (NG + 1) * 3)        // 5643

// ---------------------------------------------------------------------------
// Kernel 1: stable depth-argsort via rank counting + scatter to sorted AoS.
// ---------------------------------------------------------------------------
__global__ void sort_scatter_kernel(const float* __restrict__ means2d,
                                    const float* __restrict__ conics,
                                    const float* __restrict__ colors,
                                    const float* __restrict__ opac,
                                    const float* __restrict__ depths,
                                    float* __restrict__ ws) {
  int g = threadIdx.x;
  if (g >= NG) return;
  float d = depths[g];
  int rank = 0;
  for (int j = 0; j < NG; ++j) {
    float dj = depths[j];
    rank += (dj < d) || (dj == d && j < g);   // stable ascending (jnp.argsort)
  }
  float* p = ws + rank * PAR_STRIDE;
  p[0] = means2d[2 * g];
  p[1] = means2d[2 * g + 1];
  p[2] = conics[3 * g];
  p[3] = conics[3 * g + 1];
  p[4] = conics[3 * g + 2];
  p[5] = opac[g];
  p[6] = 0.0f;
  p[7] = 0.0f;
  ws[COL_BASE + 3 * rank]     = colors[3 * g];
  ws[COL_BASE + 3 * rank + 1] = colors[3 * g + 1];
  ws[COL_BASE + 3 * rank + 2] = colors[3 * g + 2];

  if (g == 0) {                               // dummy null gaussian at slot NG
    float* q = ws + NG * PAR_STRIDE;
    q[0] = 0.0f; q[1] = 0.0f; q[2] = 1.0f; q[3] = 0.0f;
    q[4] = 1.0f; q[5] = 0.0f; q[6] = 0.0f; q[7] = 0.0f;
    ws[COL_BASE + 3 * NG]     = 0.0f;
    ws[COL_BASE + 3 * NG + 1] = 0.0f;
    ws[COL_BASE + 3 * NG + 2] = 0.0f;
  }
}

__device__ __forceinline__ float xorswap16(float v) {
  // exact xor-16 half-wave exchange: v_permlanex16_b32 (VALU, no DS round trip)
  int r = __builtin_amdgcn_permlanex16(__float_as_int(v), __float_as_int(v),
                                       0x76543210, 0xFEDCBA98, false, false);
  return __int_as_float(r);
}

// ---------------------------------------------------------------------------
// Kernel 2: per-wave 16x1 pixel strip. Exact per-strip culling into a
// depth-ordered compact list (padded with the null gaussian), then
// 4 gaussians/iter via V_WMMA_F32_16X16X4_F32.
// ---------------------------------------------------------------------------
__launch_bounds__(256)
__global__ void raster_kernel(const float* __restrict__ ws,
                              float* __restrict__ out) {
  __shared__ float lds[WS_FLOATS];
  __shared__ unsigned short list[8 * NG];

  for (int i = threadIdx.x; i < WS_FLOATS; i += 256) lds[i] = ws[i];
  __syncthreads();

  const float4* pg4 = (const float4*)lds;   // pg4[g*2] = {mx,my,A,B}
  const float2* pg2 = (const float2*)lds;   // pg2[g*4+2] = {C,op}
  const float*  cl  = lds + COL_BASE;

  const int lane = threadIdx.x & 31;
  const int wave = threadIdx.x >> 5;
  const int pixel_base = (blockIdx.x * 8 + wave) * 16;

  const int p  = lane & 15;                 // A-matrix row M
  const int kh = lane >> 4;                 // K half (K = 2*kh, 2*kh+1)
  const float pxf = (float)((pixel_base + p) & (IMG_W - 1)) + 0.5f;
  const float yc  = (float)(pixel_base >> 8) + 0.5f;

  // ---- exact per-strip culling: keep g iff min_x sigma(x, yc) <= ln(255*op)
  const float xlo = (float)(pixel_base & (IMG_W - 1)) + 0.5f;
  const float xhi = xlo + 15.0f;
  unsigned short* mylist = list + wave * NG;
  int cnt = 0;
  for (int base = 0; base < NG; base += 32) {
    const int g = base + lane;
    float4 q0 = pg4[g * 2];                 // mx, my, A, B
    float2 q1 = pg2[g * 4 + 2];             // C, op
    float dy  = yc - q0.y;
    float dxv = -(q0.w * dy) / q0.z;        // 1-D vertex of sigma along x
    float dxc = fminf(fmaxf(dxv, xlo - q0.x), xhi - q0.x);
    float smin = 0.5f * (q0.z * dxc * dxc + q1.x * dy * dy) + q0.w * dxc * dy;
    bool pred = smin <= __logf(255.0f * q1.y);
    unsigned mask = __builtin_amdgcn_ballot_w32(pred);
    int prefix = __popc(mask & ((1u << lane) - 1u));
    if (pred) mylist[cnt + prefix] = (unsigned short)g;
    cnt += __popc(mask);
  }
  cnt = __builtin_amdgcn_readfirstlane(cnt);
  const int cntR = (cnt + 3) & ~3;
  if (lane < cntR - cnt) mylist[cnt + lane] = (unsigned short)NG;  // pad tail

  // B-matrix column for this lane (RGB in N=0..2, rest zero)
  const int   n     = lane & 15;
  const int   nc    = (n < 3) ? n : 2;
  const float nmask = (n < 3) ? 1.0f : 0.0f;

  v8f   acc = {0.f, 0.f, 0.f, 0.f, 0.f, 0.f, 0.f, 0.f};
  float T   = 1.0f;

  for (int g0 = 0; g0 < cntR; g0 += 4) {
    const int ga = g0 + 2 * kh;             // even -> one aligned b32 index load
    const unsigned pair = *(const unsigned*)(mylist + ga);
    const int ia = (int)(pair & 0xFFFFu);
    const int ib = (int)(pair >> 16);

    // alpha for (pixel p, gaussian ia)
    float4 qa = pg4[ia * 2];
    float2 ra = pg2[ia * 4 + 2];
    float dxa = pxf - qa.x, dya = yc - qa.y;
    float siga = 0.5f * (qa.z * dxa * dxa + ra.x * dya * dya) + qa.w * dxa * dya;
    float ala  = fminf(ALPHA_CLAMP, ra.y * __expf(-siga));
    ala = ((siga >= 0.0f) && (ala >= ALPHA_T)) ? ala : 0.0f;

    // alpha for (pixel p, gaussian ib)
    float4 qb = pg4[ib * 2];
    float2 rb = pg2[ib * 4 + 2];
    float dxb = pxf - qb.x, dyb = yc - qb.y;
    float sigb = 0.5f * (qb.z * dxb * dxb + rb.x * dyb * dyb) + qb.w * dxb * dyb;
    float alb  = fminf(ALPHA_CLAMP, rb.y * __expf(-sigb));
    alb = ((sigb >= 0.0f) && (alb >= ALPHA_T)) ? alb : 0.0f;

    // transmittance chain across the 4-gaussian chunk
    float oma = 1.0f - ala, omb = 1.0f - alb;
    float pp  = oma * omb;
    float ppo = xorswap16(pp);
    float pref = kh ? ppo : 1.0f;
    float Ta = T * pref;
    float Tb = Ta * oma;

    v2f a;
    a.x = (Ta >= TRANS_T) ? ala * Ta : 0.0f;
    a.y = (Tb >= TRANS_T) ? alb * Tb : 0.0f;
    T = T * pp * ppo;

    // B-matrix: rows K = 2*kh, 2*kh+1 ; column N = n (zero for N >= 3)
    v2f b;
    b.x = cl[3 * ia + nc] * nmask;
    b.y = cl[3 * ib + nc] * nmask;

    acc = __builtin_amdgcn_wmma_f32_16x16x4_f32(
        false, a, false, b, (short)0, acc, false, false);

    // wave-uniform early exit once every pixel is saturated
    if (__builtin_amdgcn_ballot_w32(T >= TRANS_T) == 0u) break;
  }

  // D layout: VGPR r holds M = r + 8*kh, column N = lane&15 (RGB only)
  if (n < 3) {
    for (int r = 0; r < 8; ++r) {
      int M = r + 8 * kh;
      out[(pixel_base + M) * 3 + n] = acc[r];
    }
  }
}

// ---------------------------------------------------------------------------
extern "C" void kernel_launch(void* const* d_in, const int* in_sizes, int n_in,
                              void* d_out, int out_size, void* d_ws, size_t ws_size,
                              hipStream_t stream) {
  (void)in_sizes; (void)n_in; (void)out_size; (void)ws_size;
  const float* means2d = (const float*)d_in[0];
  const float* conics  = (const float*)d_in[1];
  const float* colors  = (const float*)d_in[2];
  const float* opac    = (const float*)d_in[3];
  const float* depths  = (const float*)d_in[4];
  float* ws  = (float*)d_ws;
  float* out = (float*)d_out;

  sort_scatter_kernel<<<1, 512, 0, stream>>>(means2d, conics, colors, opac, depths, ws);
  raster_kernel<<<NPIX / 128, 256, 0, stream>>>(ws, out);
}